// VectorQuantizer_14181982011911
// MI455X (gfx1250) — compile-verified
//
#include <hip/hip_runtime.h>

typedef __attribute__((ext_vector_type(2))) float v2f;
typedef __attribute__((ext_vector_type(8))) float v8f;

#define NEMB   1024
#define DIM    64
#define HWSZ   4096
#define BATCH  32
#define NROWS  (BATCH * HWSZ)      // 131072 latent vectors
#define QELEMS (NROWS * DIM)       // 8388608 quantized elements
#define GATHER_BLOCKS (QELEMS / 256)  // 32768

// ---------------------------------------------------------------------------
// Kernel 1: codebook row squared-norms ||e_k||^2  (1024 values, trivial cost)
// ---------------------------------------------------------------------------
__global__ void vq_e2_kernel(const float* __restrict__ cb, float* __restrict__ e2) {
    int t = blockIdx.x * blockDim.x + threadIdx.x;
    if (t < NEMB) {
        float s = 0.0f;
        #pragma unroll
        for (int k = 0; k < DIM; ++k) { float v = cb[t * DIM + k]; s += v * v; }
        e2[t] = s;
    }
}

// ---------------------------------------------------------------------------
// Kernel 2: score GEMM + argmin via V_WMMA_F32_16X16X4_F32.
// One wave owns a 16-row tile of z_flat; loops 64 column tiles of the
// codebook; 16 chained WMMAs cover D=64 per tile. Running argmin kept per
// (row, lane) in registers, reduced cross-lane at the end.
// ---------------------------------------------------------------------------
__global__ __launch_bounds__(256) void vq_argmin_kernel(
    const float* __restrict__ z, const float* __restrict__ cb,
    const float* __restrict__ e2, float* __restrict__ idx_out) {
    const int lane = threadIdx.x & 31;
    const int wave = threadIdx.x >> 5;
    const int rowTile = blockIdx.x * 8 + wave;        // 8192 tiles total
    const int rowBase = rowTile * 16;                 // flat row n base
    const int b  = rowBase >> 12;                     // 4096 rows per batch image
    const int hw = rowBase & 4095;
    const int r  = lane & 15;                         // row (A) / column (B,C) in tile
    const int kh = lane >> 4;                         // K half selector

    // A fragment (16x4 f32 layout): a[j].x = A[r][4j+2*kh], a[j].y = +1
    // z is NCHW, so z_flat[n][k] = z[b*DIM*HW + k*HW + (n mod HW)] -> coalesced dwords
    const float* zp = z + (size_t)b * (DIM * HWSZ) + hw + r;
    v2f a[16];
    #pragma unroll
    for (int j = 0; j < 16; ++j) {
        const int k0 = 4 * j + 2 * kh;
        a[j].x = zp[(size_t)k0 * HWSZ];
        a[j].y = zp[(size_t)(k0 + 1) * HWSZ];
    }

    float minv[8];
    int   mini[8];
    #pragma unroll
    for (int i = 0; i < 8; ++i) { minv[i] = __builtin_inff(); mini[i] = 0; }

    for (int ct = 0; ct < 64; ++ct) {
        const int col = ct * 16 + r;
        const float* cp = cb + col * DIM;
        // B fragment (4x16): bf[j].x = B[2kh][col] = cb[col][4j+2kh] (aligned 8B load)
        v2f bf[16];
        #pragma unroll
        for (int j = 0; j < 16; ++j) {
            const int k0 = 4 * j + 2 * kh;
            bf[j] = *(const v2f*)(cp + k0);
        }
        const float e2v = e2[col];

        v8f c = {0.f, 0.f, 0.f, 0.f, 0.f, 0.f, 0.f, 0.f};
        #pragma unroll
        for (int j = 0; j < 16; ++j)
            c = __builtin_amdgcn_wmma_f32_16x16x4_f32(
                    /*neg_a=*/false, a[j], /*neg_b=*/false, bf[j],
                    /*c_mod=*/(short)0, c, /*reuse_a=*/false, /*reuse_b=*/false);

        // C layout: VGPR i holds row (i + 8*kh), column = lane%16
        #pragma unroll
        for (int i = 0; i < 8; ++i) {
            const float s = e2v - 2.0f * c[i];
            if (s < minv[i]) { minv[i] = s; mini[i] = col; }
        }
    }

    // Cross-lane argmin within each 16-lane half (wave32): xor masks 8,4,2,1.
    // Tie-break to smaller index to match jnp.argmin.
    #pragma unroll
    for (int off = 8; off >= 1; off >>= 1) {
        #pragma unroll
        for (int i = 0; i < 8; ++i) {
            const float ov = __shfl_xor(minv[i], off, 32);
            const int   oi = __shfl_xor(mini[i], off, 32);
            if (ov < minv[i] || (ov == minv[i] && oi < mini[i])) {
                minv[i] = ov; mini[i] = oi;
            }
        }
    }

    if (r == 0) {
        const int rowOff = rowBase + kh * 8;
        #pragma unroll
        for (int i = 0; i < 8; ++i)
            idx_out[rowOff + i] = (float)mini[i];
    }
}

// ---------------------------------------------------------------------------
// Kernel 3: gather quantized output (NCHW) + per-block partial of (q-z)^2
// ---------------------------------------------------------------------------
__global__ __launch_bounds__(256) void vq_gather_kernel(
    const float* __restrict__ z, const float* __restrict__ cb,
    const float* __restrict__ idx_f, float* __restrict__ quant,
    float* __restrict__ partials) {
    __shared__ float red[256];
    const size_t t = (size_t)blockIdx.x * 256 + threadIdx.x;
    const int hw = (int)(t & 4095);
    const int c  = (int)((t >> 12) & 63);
    const int b  = (int)(t >> 18);
    const int n  = b * HWSZ + hw;
    const int k  = (int)idx_f[n];
    const float q = cb[(size_t)k * DIM + c];
    const float d = q - z[t];
    quant[t] = q;
    red[threadIdx.x] = d * d;
    __syncthreads();
    for (int s = 128; s > 0; s >>= 1) {
        if ((int)threadIdx.x < s) red[threadIdx.x] += red[threadIdx.x + s];
        __syncthreads();
    }
    if (threadIdx.x == 0) partials[blockIdx.x] = red[0];
}

// ---------------------------------------------------------------------------
// Kernel 4: final loss = (1 + 0.25) * mean((q - z)^2)   (forward values of
// q_latent_loss and e_latent_loss are identical; stop_gradient is a no-op fwd)
// ---------------------------------------------------------------------------
__global__ __launch_bounds__(256) void vq_loss_kernel(
    const float* __restrict__ partials, float* __restrict__ loss) {
    __shared__ float red[256];
    float s = 0.0f;
    for (int i = threadIdx.x; i < GATHER_BLOCKS; i += 256) s += partials[i];
    red[threadIdx.x] = s;
    __syncthreads();
    for (int st = 128; st > 0; st >>= 1) {
        if ((int)threadIdx.x < st) red[threadIdx.x] += red[threadIdx.x + st];
        __syncthreads();
    }
    if (threadIdx.x == 0) loss[0] = 1.25f * red[0] / (float)QELEMS;
}

// ---------------------------------------------------------------------------
// d_out layout (tuple, flat, return order): quantized [8388608] | loss [1] |
// indices [131072] (written as float; dtype of d_out is float).
// d_ws: e2 [1024 f] | partials [32768 f]  (~132 KB)
// ---------------------------------------------------------------------------
extern "C" void kernel_launch(void* const* d_in, const int* in_sizes, int n_in,
                              void* d_out, int out_size, void* d_ws, size_t ws_size,
                              hipStream_t stream) {
    const float* z  = (const float*)d_in[0];
    const float* cb = (const float*)d_in[1];

    float* quant = (float*)d_out;
    float* loss  = quant + QELEMS;
    float* idxf  = loss + 1;

    float* e2       = (float*)d_ws;
    float* partials = e2 + NEMB;

    vq_e2_kernel<<<(NEMB + 255) / 256, 256, 0, stream>>>(cb, e2);
    vq_argmin_kernel<<<NROWS / (16 * 8), 256, 0, stream>>>(z, cb, e2, idxf);
    vq_gather_kernel<<<GATHER_BLOCKS, 256, 0, stream>>>(z, cb, idxf, quant, partials);
    vq_loss_kernel<<<1, 256, 0, stream>>>(partials, loss);
}